// GNNLoRA_84018150244514
// MI455X (gfx1250) — compile-verified
//
#include <hip/hip_runtime.h>
#include <math.h>

// ---------------------------------------------------------------------------
// Types for WMMA (CDNA5 / gfx1250, wave32)
// ---------------------------------------------------------------------------
typedef __attribute__((ext_vector_type(2))) float v2f;
typedef __attribute__((ext_vector_type(8))) float v8f;

#define NEG_SLOPE 0.2f

// Order-preserving float -> uint mapping so atomicMax(uint) == float max.
__device__ __forceinline__ unsigned f2ord(float f) {
    unsigned u = __float_as_uint(f);
    return (u & 0x80000000u) ? ~u : (u | 0x80000000u);
}
__device__ __forceinline__ float ord2f(unsigned u) {
    return (u & 0x80000000u) ? __uint_as_float(u & 0x7fffffffu)
                             : __uint_as_float(~u);
}

// ---------------------------------------------------------------------------
// FP32 WMMA GEMM: C[nrows x CN] = X[nrows x K] @ W[K x CN], row-major.
// One wave computes a 16-row stripe across all CN columns using
// V_WMMA_F32_16X16X4_F32. nrows % 16 == 0 (N=50000=3125*16), K%4==0, CN%16==0.
//
// A (16x4, fp32, 2 VGPRs): lane = m + 16*(krel>>1), vgpr = krel&1
// B (4x16, fp32, 2 VGPRs): vgpr0 = rows {k0, k0+2}, vgpr1 = rows {k0+1, k0+3}
//                          (lanes 0-15 -> first row of pair, 16-31 -> second)
// C (16x16, fp32, 8 VGPRs): vgpr r = rows {r, r+8}, col = lane&15
// ---------------------------------------------------------------------------
template <int K, int CN>
__global__ void gemm_f32_wmma(const float* __restrict__ X,
                              const float* __restrict__ W,
                              float* __restrict__ C, int nrows) {
    const int lane  = threadIdx.x & 31;
    const int wave  = threadIdx.x >> 5;
    const int tile  = blockIdx.x * (blockDim.x >> 5) + wave;
    if (tile * 16 >= nrows) return;       // wave-uniform guard: EXEC stays all-1s

    const int row0   = tile * 16;
    const int m      = lane & 15;         // row (A,C) / col (B) within tile
    const int khalf  = lane >> 4;         // 0: k{0,1}, 1: k{2,3}
    constexpr int NT = CN / 16;

    v8f acc[NT] = {};                     // zero accumulators

    const float* __restrict__ xrow = X + (size_t)(row0 + m) * K;

    for (int k0 = 0; k0 < K; k0 += 4) {
        const int ka = k0 + khalf * 2;
        v2f a;
        a.x = xrow[ka];
        a.y = xrow[ka + 1];
#pragma unroll
        for (int t = 0; t < NT; ++t) {
            const int col = t * 16 + m;
            v2f b;
            b.x = W[(size_t)ka * CN + col];
            b.y = W[(size_t)(ka + 1) * CN + col];
            acc[t] = __builtin_amdgcn_wmma_f32_16x16x4_f32(
                false, a, false, b, (short)0, acc[t], false, false);
        }
    }

#pragma unroll
    for (int t = 0; t < NT; ++t) {
#pragma unroll
        for (int r = 0; r < 8; ++r) {
            const int row = row0 + r + 8 * khalf;
            C[(size_t)row * CN + t * 16 + m] = acc[t][r];
        }
    }
}

// ---------------------------------------------------------------------------
// GAT layer kernels (C = 128 channels, H = 1 head)
// ---------------------------------------------------------------------------

// One wave per node: a_s[n]=dot(hp[n],att_s), a_d[n]=dot(hp[n],att_d),
// init segment-max (ordered -inf) / segment-sum (0), zero output row.
__global__ void node_prep_kernel(const float* __restrict__ hp,
                                 const float* __restrict__ att_s,
                                 const float* __restrict__ att_d,
                                 float* __restrict__ a_s, float* __restrict__ a_d,
                                 unsigned* __restrict__ m_ord,
                                 float* __restrict__ den,
                                 float* __restrict__ out, int N) {
    const int wid  = (blockIdx.x * blockDim.x + threadIdx.x) >> 5;
    const int lane = threadIdx.x & 31;
    if (wid >= N) return;

    const float4 h = ((const float4*)(hp + (size_t)wid * 128))[lane];
    const float4 s = ((const float4*)att_s)[lane];
    const float4 d = ((const float4*)att_d)[lane];
    float ds = h.x * s.x + h.y * s.y + h.z * s.z + h.w * s.w;
    float dd = h.x * d.x + h.y * d.y + h.z * d.z + h.w * d.w;
#pragma unroll
    for (int o = 16; o > 0; o >>= 1) {
        ds += __shfl_xor(ds, o, 32);
        dd += __shfl_xor(dd, o, 32);
    }
    if (lane == 0) {
        a_s[wid]   = ds;
        a_d[wid]   = dd;
        m_ord[wid] = f2ord(-INFINITY);
        den[wid]   = 0.0f;
    }
    ((float4*)(out + (size_t)wid * 128))[lane] = make_float4(0.f, 0.f, 0.f, 0.f);
}

// One thread per edge (E real edges + N self loops appended).
__global__ void edge_max_kernel(const int* __restrict__ src,
                                const int* __restrict__ dst,
                                const float* __restrict__ a_s,
                                const float* __restrict__ a_d,
                                unsigned* __restrict__ m_ord,
                                float* __restrict__ ebuf, int E, int N) {
    const int i = blockIdx.x * blockDim.x + threadIdx.x;
    if (i >= E + N) return;
    const int s = (i < E) ? src[i] : (i - E);
    const int d = (i < E) ? dst[i] : (i - E);
    float e = a_s[s] + a_d[d];
    e = (e > 0.0f) ? e : NEG_SLOPE * e;   // leaky relu
    ebuf[i] = e;
    atomicMax(m_ord + d, f2ord(e));
}

__global__ void edge_sum_kernel(const int* __restrict__ src,
                                const int* __restrict__ dst,
                                const unsigned* __restrict__ m_ord,
                                float* __restrict__ den,
                                float* __restrict__ ebuf, int E, int N) {
    const int i = blockIdx.x * blockDim.x + threadIdx.x;
    if (i >= E + N) return;
    const int d  = (i < E) ? dst[i] : (i - E);
    const float ex = __expf(ebuf[i] - ord2f(m_ord[d]));
    ebuf[i] = ex;
    atomicAdd(den + d, ex);
}

// One wave per edge: out[dst] += hp[src] * alpha  (float4 per lane -> 128 ch).
__global__ void edge_scatter_kernel(const int* __restrict__ src,
                                    const int* __restrict__ dst,
                                    const float* __restrict__ hp,
                                    const float* __restrict__ ebuf,
                                    const float* __restrict__ den,
                                    float* __restrict__ out, int E, int N) {
    const int wid  = (blockIdx.x * blockDim.x + threadIdx.x) >> 5;
    const int lane = threadIdx.x & 31;
    if (wid >= E + N) return;
    const int s = (wid < E) ? src[wid] : (wid - E);
    const int d = (wid < E) ? dst[wid] : (wid - E);
    const float alpha = ebuf[wid] / (den[d] + 1e-16f);

    const float4 v = ((const float4*)(hp + (size_t)s * 128))[lane];
    float* op = out + (size_t)d * 128 + lane * 4;
    atomicAdd(op + 0, v.x * alpha);
    atomicAdd(op + 1, v.y * alpha);
    atomicAdd(op + 2, v.z * alpha);
    atomicAdd(op + 3, v.w * alpha);
}

// out += bias (head mean is identity for H=1); optional relu.
__global__ void finalize_kernel(float* __restrict__ out,
                                const float* __restrict__ bias,
                                int total, int do_relu) {
    const int i = blockIdx.x * blockDim.x + threadIdx.x;
    if (i >= total) return;
    float v = out[i] + bias[i & 127];
    if (do_relu) v = fmaxf(v, 0.0f);
    out[i] = v;
}

__global__ void fuse_kernel(const float* __restrict__ a,
                            const float* __restrict__ b,
                            float* __restrict__ c, int total) {
    const int i = blockIdx.x * blockDim.x + threadIdx.x;
    if (i >= total) return;
    c[i] = a[i] + b[i];
}

// ---------------------------------------------------------------------------
// Host-side orchestration
// ---------------------------------------------------------------------------
static void run_gat_layer(const float* hp, const float* att_s, const float* att_d,
                          const float* bias, float* out, int do_relu,
                          const int* src, const int* dst, int E, int N,
                          float* a_s, float* a_d, unsigned* m_ord, float* den,
                          float* ebuf, hipStream_t stream) {
    const int Etot = E + N;
    const int TB = 256;
    // prep: wave per node
    node_prep_kernel<<<(N * 32 + TB - 1) / TB, TB, 0, stream>>>(
        hp, att_s, att_d, a_s, a_d, m_ord, den, out, N);
    // segment max
    edge_max_kernel<<<(Etot + TB - 1) / TB, TB, 0, stream>>>(
        src, dst, a_s, a_d, m_ord, ebuf, E, N);
    // exp + segment sum
    edge_sum_kernel<<<(Etot + TB - 1) / TB, TB, 0, stream>>>(
        src, dst, m_ord, den, ebuf, E, N);
    // weighted scatter: wave per edge
    edge_scatter_kernel<<<((size_t)Etot * 32 + TB - 1) / TB, TB, 0, stream>>>(
        src, dst, hp, ebuf, den, out, E, N);
    // bias (+relu)
    finalize_kernel<<<(N * 128 + TB - 1) / TB, TB, 0, stream>>>(
        out, bias, N * 128, do_relu);
}

template <int K, int CN>
static void run_gemm(const float* X, const float* W, float* C, int nrows,
                     hipStream_t stream) {
    const int tiles = (nrows + 15) / 16;
    const int waves_per_block = 4;                 // 128 threads
    const int blocks = (tiles + waves_per_block - 1) / waves_per_block;
    gemm_f32_wmma<K, CN><<<blocks, waves_per_block * 32, 0, stream>>>(X, W, C, nrows);
}

extern "C" void kernel_launch(void* const* d_in, const int* in_sizes, int n_in,
                              void* d_out, int out_size, void* d_ws, size_t ws_size,
                              hipStream_t stream) {
    const float* x        = (const float*)d_in[0];
    const int*   ei       = (const int*)d_in[1];
    const float* W1       = (const float*)d_in[2];
    const float* att_src1 = (const float*)d_in[3];
    const float* att_dst1 = (const float*)d_in[4];
    const float* b1       = (const float*)d_in[5];
    const float* W2       = (const float*)d_in[6];
    const float* att_src2 = (const float*)d_in[7];
    const float* att_dst2 = (const float*)d_in[8];
    const float* b2       = (const float*)d_in[9];
    const float* A1       = (const float*)d_in[10];
    const float* B1       = (const float*)d_in[11];
    const float* latt_s1  = (const float*)d_in[12];
    const float* latt_d1  = (const float*)d_in[13];
    const float* lb1      = (const float*)d_in[14];
    const float* A2       = (const float*)d_in[15];
    const float* B2       = (const float*)d_in[16];
    const float* latt_s2  = (const float*)d_in[17];
    const float* latt_d2  = (const float*)d_in[18];
    const float* lb2      = (const float*)d_in[19];

    const int N = in_sizes[0] / 128;   // 50000
    const int E = in_sizes[1] / 2;     // 800000
    const int* src = ei;               // edge_index[0]
    const int* dst = ei + E;           // edge_index[1]

    const size_t NC = (size_t)N * 128;

    // ---- workspace layout (floats) ----
    float*    ws    = (float*)d_ws;
    float*    buf0  = ws;                         // N*128  projected features
    float*    buf1  = buf0 + NC;                  // N*128  layer outputs
    float*    bufS  = buf1 + NC;                  // N*32   LoRA rank buffer
    float*    a_s   = bufS + (size_t)N * 32;      // N
    float*    a_d   = a_s + N;                    // N
    unsigned* m_ord = (unsigned*)(a_d + N);       // N
    float*    den   = (float*)(m_ord + N);        // N
    float*    ebuf  = den + N;                    // E + N

    float* out_fused = (float*)d_out;
    float* out_main  = out_fused + NC;
    float* out_lora  = out_fused + 2 * NC;

    // ---- main branch: GAT(W1) -> relu -> GAT(W2) ----
    run_gemm<128, 128>(x, W1, buf0, N, stream);                         // h1p = x@W1
    run_gat_layer(buf0, att_src1, att_dst1, b1, buf1, /*relu=*/1,
                  src, dst, E, N, a_s, a_d, m_ord, den, ebuf, stream);  // h = relu(gat1)
    run_gemm<128, 128>(buf1, W2, buf0, N, stream);                      // h2p = h@W2
    run_gat_layer(buf0, att_src2, att_dst2, b2, out_main, /*relu=*/0,
                  src, dst, E, N, a_s, a_d, m_ord, den, ebuf, stream);  // main

    // ---- LoRA branch: GAT((x@A1)@B1) -> GAT((g@A2)@B2) ----
    run_gemm<128, 32>(x, A1, bufS, N, stream);                          // xA = x@A1
    run_gemm<32, 128>(bufS, B1, buf0, N, stream);                       // g1p = xA@B1
    run_gat_layer(buf0, latt_s1, latt_d1, lb1, buf1, /*relu=*/0,
                  src, dst, E, N, a_s, a_d, m_ord, den, ebuf, stream);  // g
    run_gemm<128, 32>(buf1, A2, bufS, N, stream);                       // gA = g@A2
    run_gemm<32, 128>(bufS, B2, buf0, N, stream);                       // g2p = gA@B2
    run_gat_layer(buf0, latt_s2, latt_d2, lb2, out_lora, /*relu=*/0,
                  src, dst, E, N, a_s, a_d, m_ord, den, ebuf, stream);  // lora

    // ---- fused = main + lora ----
    fuse_kernel<<<((int)NC + 255) / 256, 256, 0, stream>>>(out_main, out_lora,
                                                           out_fused, (int)NC);
}